// Vad_86242943303722
// MI455X (gfx1250) — compile-verified
//
#include <hip/hip_runtime.h>
#include <math.h>

typedef float v2f __attribute__((ext_vector_type(2)));
typedef float v4f __attribute__((ext_vector_type(4)));
typedef float v8f __attribute__((ext_vector_type(8)));

// ---- problem constants (sample_rate = 16000 baked in) ----
#define T_SAMP   16000000
#define MLEN     1600      // measure_len
#define NDFT     2048
#define PERIOD   800
#define KMEAS    19999     // (T - MLEN)/PERIOD + 1
#define KPAD     20000     // KMEAS padded to x16
#define SPEC_S   6         // spec_start
#define SBINS    762       // spec_end - spec_start
#define SPAD     768       // SBINS padded to x16
#define CEP_S    4
#define NCEP     49        // cep_end - cep_start
#define NCPAD    64
#define NDFT2    1024      // cepstrum rfft length

#define MEAS_MULT 0.8824969025845955f    // exp(-1/(0.4*20))
#define UP_MULT   0.6065306597126334f    // exp(-1/(0.1*20))
#define DOWN_MULT 0.006737946999085467f  // exp(-1/(0.01*20))
#define TRIG_MULT 0.8187307530779818f    // exp(-1/(0.25*20))
#define NOISE_RED 1.35f
#define TRIG_LVL  7.0f
#define BOOT_MAX  6
#define NRING     20
#define GAPL      5
#define TWO_PI    6.2831853071795864f

// (-2.0 / -2^31) / sqrt(1600)  — SoX int32 scale, forces fp32 math
#define CS_SCALE  2.3283064365386963e-11f

// ---- workspace layout (float offsets) ----
// Packed spectrum basis: [MLEN/4][2][SPAD] float4 = {wcos(r), wcos(r+1), wsin(r), wsin(r+1)}
static constexpr size_t OFF_BP   = 0;                               // 2,457,600 floats
static constexpr size_t OFF_BP2  = OFF_BP  + (size_t)MLEN * SPAD * 2 / 2;  // = 1,228,800
// Packed cepstrum basis: [SPAD/4][2][NCPAD] float4                  // 98,304 floats
static constexpr size_t OFF_D    = OFF_BP2 + (size_t)SPAD * NCPAD * 2;     // d -> dd in place [KPAD x SPAD]
static constexpr size_t OFF_CEP2 = OFF_D   + (size_t)KPAD * SPAD;          // [KPAD x NCPAD]

// ---------------------------------------------------------------------------
// Spectrum basis, fragment-ready: one float4 per (k-step, half, col):
//   {w[r]*cos(th(r)), w[r+1]*cos(th(r+1)), w[r]*sin(th(r)), w[r+1]*sin(th(r+1))}
// th(n) = 2*pi*((6+col)*n mod 2048)/2048  (integer mod keeps phase exact).
// ---------------------------------------------------------------------------
__global__ void vad_init_spec_basis(v4f* __restrict__ bp) {
  int idx = blockIdx.x * blockDim.x + threadIdx.x;
  if (idx >= (MLEN / 4) * 2 * SPAD) return;
  int col  = idx % SPAD;
  int rest = idx / SPAD;
  int half = rest & 1;
  int r0   = (rest >> 1) * 4 + half * 2;
  v4f v = {0.0f, 0.0f, 0.0f, 0.0f};
  if (col < SBINS) {
    int j = SPEC_S + col;
#pragma unroll
    for (int q = 0; q < 2; ++q) {
      int n = r0 + q;
      float wn = CS_SCALE * 0.5f * (1.0f - cosf(TWO_PI * (float)n / (float)MLEN));
      int m = (j * n) & (NDFT - 1);
      float ang = (float)m * (TWO_PI / (float)NDFT);
      v[q]     = wn * cosf(ang);
      v[q + 2] = wn * sinf(ang);
    }
  }
  bp[idx] = v;
}

// Cepstrum basis, fragment-ready: {cos(r), cos(r+1), sin(r), sin(r+1)} per float4.
__global__ void vad_init_cep_basis(v4f* __restrict__ bp2) {
  int idx = blockIdx.x * blockDim.x + threadIdx.x;
  if (idx >= (SPAD / 4) * 2 * NCPAD) return;
  int col  = idx % NCPAD;
  int rest = idx / NCPAD;
  int half = rest & 1;
  int r0   = (rest >> 1) * 4 + half * 2;
  v4f v = {0.0f, 0.0f, 0.0f, 0.0f};
  if (col < NCEP) {
    int j = CEP_S + col;
#pragma unroll
    for (int q = 0; q < 2; ++q) {
      int s = r0 + q;
      if (s < SBINS) {
        int m = (j * (SPEC_S + s)) & (NDFT2 - 1);
        float ang = (float)m * (TWO_PI / (float)NDFT2);
        v[q]     = cosf(ang);
        v[q + 2] = sinf(ang);
      }
    }
  }
  bp2[idx] = v;
}

// ---------------------------------------------------------------------------
// Phase 1: spectrum magnitudes via fp32 WMMA GEMM (window folded into basis).
// Per 4-deep k-step: 1x global_load_b64 (A pair) + 1x global_load_b128
// (packed cos/sin B fragments) + 2x v_wmma_f32_16x16x4_f32.
// ---------------------------------------------------------------------------
__global__ __launch_bounds__(256) void vad_spectrum_wmma(
    const float* __restrict__ wf, const v4f* __restrict__ bp,
    float* __restrict__ dmat) {
  const int NT = SPAD / 16;  // 48 column tiles
  int wave = (blockIdx.x * 256 + (int)threadIdx.x) >> 5;  // 0..59999
  int tm = wave / NT, tn = wave % NT;
  int lane = threadIdx.x & 31;
  int half = lane >> 4, lm = lane & 15;

  int krow = tm * 16 + lm;
  if (krow >= KMEAS) krow = KMEAS - 1;  // duplicate last row into pad
  const float* ap = wf + (size_t)krow * PERIOD + half * 2;
  int col = tn * 16 + lm;
  const v4f* bpp = bp + (size_t)half * SPAD + col;

  v8f accC = {};
  v8f accS = {};
#pragma unroll 4
  for (int kk = 0; kk < MLEN; kk += 4) {
    v2f a = *(const v2f*)(ap + kk);
    v4f b = bpp[(size_t)(kk >> 2) * (2 * SPAD)];
    v2f vbc; vbc.x = b.x; vbc.y = b.y;
    v2f vbs; vbs.x = b.z; vbs.y = b.w;
    accC = __builtin_amdgcn_wmma_f32_16x16x4_f32(false, a, false, vbc, (short)0, accC, false, false);
    accS = __builtin_amdgcn_wmma_f32_16x16x4_f32(false, a, false, vbs, (short)0, accS, false, false);
  }
  int rbase = tm * 16 + half * 8;
  float* drow = dmat + (size_t)rbase * SPAD + col;
#pragma unroll
  for (int i = 0; i < 8; ++i) {
    float re = accC[i], im = accS[i];
    drow[(size_t)i * SPAD] = __builtin_amdgcn_sqrtf(re * re + im * im);
  }
}

// ---------------------------------------------------------------------------
// Phase 2: per-bin sequential noise gate (the only truly serial part).
// In-place: reads d[k][s], writes dd[k][s]*cw[s]. 768 independent chains.
// Padding bins naturally stay 0 (zero basis cols -> d=0 -> dd=0).
// ---------------------------------------------------------------------------
__global__ __launch_bounds__(128) void vad_noise_gate(float* __restrict__ dmat) {
  int s = blockIdx.x * 128 + threadIdx.x;  // 0..767
  float cwv = (2.0f / sqrtf((float)SBINS)) * 0.5f *
              (1.0f - cosf(TWO_PI * (float)s / (float)SBINS));
  float spec = 0.0f, noise = 0.0f;
  float* p = dmat + s;
  for (int k = 0; k < KPAD; ++k) {
    float d = p[(size_t)k * SPAD];
    if (k + 16 < KPAD) __builtin_prefetch(p + (size_t)(k + 16) * SPAD, 0, 0);
    float mult = (k <= BOOT_MAX) ? ((float)k / (float)(k + 1)) : MEAS_MULT;
    spec = spec * mult + d * (1.0f - mult);
    float d2 = spec * spec;
    float nm = (k <= BOOT_MAX) ? 0.0f : ((d2 > noise) ? UP_MULT : DOWN_MULT);
    noise = noise * nm + d2 * (1.0f - nm);
    float dd = __builtin_amdgcn_sqrtf(fmaxf(0.0f, d2 - NOISE_RED * noise)) * cwv;
    p[(size_t)k * SPAD] = dd;
  }
}

// ---------------------------------------------------------------------------
// Phase 3: cepstrum power via fp32 WMMA GEMM: [KPAD x SPAD] x [SPAD x 64].
// Same packed-basis trick; epilogue writes re^2+im^2 into CEP2.
// ---------------------------------------------------------------------------
__global__ __launch_bounds__(256) void vad_cepstrum_wmma(
    const float* __restrict__ dmat, const v4f* __restrict__ bp2,
    float* __restrict__ cep2) {
  const int NT = NCPAD / 16;  // 4 column tiles
  int wave = (blockIdx.x * 256 + (int)threadIdx.x) >> 5;  // 0..4999
  int tm = wave / NT, tn = wave % NT;
  int lane = threadIdx.x & 31;
  int half = lane >> 4, lm = lane & 15;

  const float* ap = dmat + (size_t)(tm * 16 + lm) * SPAD + half * 2;
  int col = tn * 16 + lm;
  const v4f* bpp = bp2 + (size_t)half * NCPAD + col;

  v8f accC = {};
  v8f accS = {};
#pragma unroll 4
  for (int kk = 0; kk < SPAD; kk += 4) {
    v2f a = *(const v2f*)(ap + kk);
    v4f b = bpp[(size_t)(kk >> 2) * (2 * NCPAD)];
    v2f vbc; vbc.x = b.x; vbc.y = b.y;
    v2f vbs; vbs.x = b.z; vbs.y = b.w;
    accC = __builtin_amdgcn_wmma_f32_16x16x4_f32(false, a, false, vbc, (short)0, accC, false, false);
    accS = __builtin_amdgcn_wmma_f32_16x16x4_f32(false, a, false, vbs, (short)0, accS, false, false);
  }
  int rbase = tm * 16 + half * 8;
  float* crow = cep2 + (size_t)rbase * NCPAD + col;
#pragma unroll
  for (int i = 0; i < 8; ++i) {
    float re = accC[i], im = accS[i];
    crow[(size_t)i * NCPAD] = re * re + im * im;
  }
}

// Phase 4: per-measure reduce + log -> meas_all into d_out[0..K)
__global__ void vad_measure(const float* __restrict__ cep2, float* __restrict__ out) {
  int k = blockIdx.x * blockDim.x + threadIdx.x;
  if (k >= KMEAS) return;
  const v4f* r = (const v4f*)(cep2 + (size_t)k * NCPAD);
  float s = 0.0f;
#pragma unroll
  for (int c = 0; c < NCPAD / 4; ++c) {
    v4f v = r[c];  // padding columns are exactly 0
    s += v.x + v.y + v.z + v.w;
  }
  float meas = 0.0f;
  if (s > 0.0f) {
    float v = 21.0f + logf(fmaxf(s, 1.17549435e-38f) * (1.0f / (float)NCEP));
    meas = fmaxf(0.0f, v);
  }
  out[k] = meas;
}

// Phase 5: scalar EMA (mean_all), trigger detection, backward ring search.
__global__ void vad_finalize(float* __restrict__ out) {
  const float* meas = out;
  float* mean = out + KMEAS;
  float m = 0.0f;
  int t = -1;
  for (int k = 0; k < KMEAS; ++k) {
    m = m * TRIG_MULT + meas[k] * (1.0f - TRIG_MULT);
    mean[k] = m;
    if (t < 0 && m >= TRIG_LVL) t = k;
  }
  float trig_f, start_f;
  if (t < 0) {
    trig_f = -1.0f;
    start_f = (float)T_SAMP;
  } else {
    int jT = NRING, jZ = NRING;
    for (int j = 0; j < NRING; ++j) {
      int idx = t - j;
      float mj = (idx >= 0) ? meas[idx] : 0.0f;
      bool hit = (mj >= TRIG_LVL) && (j <= jT + GAPL);
      bool zero = (mj == 0.0f) && (jT >= jZ);
      int jZn = hit ? j : (zero ? j : jZ);
      jT = hit ? j : jT;
      jZ = jZn;
    }
    int nf = jZ;
    if (nf > NRING - 1) nf = NRING - 1;
    if (nf < 0) nf = 0;
    long os = (long)(t - nf) * (long)PERIOD;  // fixed_pre == 0
    if (os < 0) os = 0;
    if (os > T_SAMP) os = T_SAMP;
    trig_f = (float)t;
    start_f = (float)os;
  }
  out[2 * KMEAS] = trig_f;
  out[2 * KMEAS + 1] = start_f;
}

extern "C" void kernel_launch(void* const* d_in, const int* in_sizes, int n_in,
                              void* d_out, int out_size, void* d_ws, size_t ws_size,
                              hipStream_t stream) {
  (void)in_sizes; (void)n_in; (void)out_size; (void)ws_size;
  const float* wf = (const float*)d_in[0];
  float* ws   = (float*)d_ws;
  v4f*  bp    = (v4f*)(ws + OFF_BP);
  v4f*  bp2   = (v4f*)(ws + OFF_BP2);
  float* dmat = ws + OFF_D;
  float* cep2 = ws + OFF_CEP2;
  float* out  = (float*)d_out;

  vad_init_spec_basis<<<((MLEN / 4) * 2 * SPAD + 255) / 256, 256, 0, stream>>>(bp);
  vad_init_cep_basis<<<((SPAD / 4) * 2 * NCPAD + 255) / 256, 256, 0, stream>>>(bp2);
  // 1250 row tiles * 48 col tiles = 60000 waves = 7500 blocks of 8 waves
  vad_spectrum_wmma<<<7500, 256, 0, stream>>>(wf, bp, dmat);
  // 768 independent sequential chains across 6 WGPs
  vad_noise_gate<<<6, 128, 0, stream>>>(dmat);
  // 1250 * 4 = 5000 waves = 625 blocks
  vad_cepstrum_wmma<<<625, 256, 0, stream>>>(dmat, bp2, cep2);
  vad_measure<<<(KMEAS + 255) / 256, 256, 0, stream>>>(cep2, out);
  vad_finalize<<<1, 1, 0, stream>>>(out);
}